// GeneRegulatoryNetwork_25185688224017
// MI455X (gfx1250) — compile-verified
//
#include <hip/hip_runtime.h>

// GeneRegulatoryNetwork fused kernel for gfx1250 (MI455X).
// out[i,j] = sign(argmax(MLP(cat[x_i,x_j]))) * (x_i^T Wb x_j + bb), zero diag.
//
// Key optimization vs reference: never materialize the [768,768,128] hidden
// tensor (300MB of HBM traffic). Everything is fused per 16x16 output tile.
// Affinity GEMM (768x768x128 f32) runs on V_WMMA_F32_16X16X4_F32.

#define N_GENES 768
#define HID     128
#define NTILE   48   // 768 / 16

typedef float v2f __attribute__((ext_vector_type(2)));
typedef float v8f __attribute__((ext_vector_type(8)));

// ---------------------------------------------------------------------------
// Kernel 1: three per-gene projections into workspace.
//   Ai[i,c] = sum_k X[i,k] * W1[c, k]        (W1 is [128, 256] row-major)
//   Bj[i,c] = sum_k X[i,k] * W1[c, 128+k]
//   Y [i,c] = sum_k X[i,k] * Wb[k, c]        (Wb is [128, 128] row-major)
// One block per gene row, one thread per output channel.
// ---------------------------------------------------------------------------
__global__ __launch_bounds__(HID) void grn_proj_kernel(
    const float* __restrict__ X, const float* __restrict__ W1,
    const float* __restrict__ Wb,
    float* __restrict__ Ai, float* __restrict__ Bj, float* __restrict__ Y) {
  __shared__ float sX[HID];
  const int i = blockIdx.x;
  const int c = threadIdx.x;
  sX[c] = X[i * HID + c];
  __syncthreads();
  const float* __restrict__ w1a = W1 + c * (2 * HID);
  const float* __restrict__ w1b = w1a + HID;
  float a = 0.f, b = 0.f, y = 0.f;
#pragma unroll 8
  for (int k = 0; k < HID; ++k) {
    const float xv = sX[k];
    a = fmaf(xv, w1a[k], a);
    b = fmaf(xv, w1b[k], b);
    y = fmaf(xv, Wb[k * HID + c], y);  // coalesced across lanes (fixed k)
  }
  Ai[i * HID + c] = a;
  Bj[i * HID + c] = b;
  Y[i * HID + c]  = y;
}

// ---------------------------------------------------------------------------
// Kernel 2: one wave32 per 16x16 output tile (EXEC all ones -> WMMA legal).
//   (a) affinity tile = Y[i0:i0+16,:] x X[j0:j0+16,:]^T + bb  via
//       32 chained V_WMMA_F32_16X16X4_F32 (K = 128 in steps of 4).
//   (b) classifier sign per pair via VALU from LDS-staged Ai/Bj/b1/W2.
//   (c) combine, zero diagonal, store.
// ---------------------------------------------------------------------------
__global__ __launch_bounds__(32) void grn_tile_kernel(
    const float* __restrict__ X,  const float* __restrict__ Ai,
    const float* __restrict__ Bj, const float* __restrict__ Y,
    const float* __restrict__ b1, const float* __restrict__ W2,
    const float* __restrict__ b2, const float* __restrict__ bb,
    float* __restrict__ out) {
  __shared__ alignas(16) float sAi[16 * HID];
  __shared__ alignas(16) float sBj[16 * HID];
  __shared__ alignas(16) float sB1[HID];
  __shared__ alignas(16) float sW2[3 * HID];
  __shared__ float sSign[256];

  const int lane = threadIdx.x;        // 0..31
  const int i0 = blockIdx.x * 16;
  const int j0 = blockIdx.y * 16;

  // --- cooperative LDS staging (float4 = global_load_b128 / ds_store_b128) ---
  {
    const float4* __restrict__ gAi = (const float4*)(Ai + i0 * HID);
    const float4* __restrict__ gBj = (const float4*)(Bj + j0 * HID);
    float4* dAi = (float4*)sAi;
    float4* dBj = (float4*)sBj;
#pragma unroll
    for (int q = 0; q < 16; ++q) {     // 16 * 32 lanes = 512 float4 per slab
      dAi[lane + 32 * q] = gAi[lane + 32 * q];
      dBj[lane + 32 * q] = gBj[lane + 32 * q];
    }
    ((float4*)sB1)[lane] = ((const float4*)b1)[lane];  // 32 float4 = 128 f
#pragma unroll
    for (int q = 0; q < 3; ++q)
      ((float4*)sW2)[lane + 32 * q] = ((const float4*)W2)[lane + 32 * q];
  }
  __syncthreads();

  // --- (a) affinity tile on the WMMA pipe ---------------------------------
  // A frag (16x4 f32): lane L holds row M = L&15, K pair base = 2*(L>>4).
  // B frag (4x16 f32): same lane->(row,Kpair) striping, rows taken from X.
  const float bbv = bb[0];
  v8f acc = {bbv, bbv, bbv, bbv, bbv, bbv, bbv, bbv};
  const int lm = lane & 15;
  const int kb = (lane >> 4) << 1;  // 0 or 2
  const float* __restrict__ arow = Y + (i0 + lm) * HID + kb;
  const float* __restrict__ brow = X + (j0 + lm) * HID + kb;
#pragma unroll
  for (int kk = 0; kk < HID; kk += 4) {
    const v2f afrag = *(const v2f*)(arow + kk);
    const v2f bfrag = *(const v2f*)(brow + kk);
    acc = __builtin_amdgcn_wmma_f32_16x16x4_f32(
        /*neg_a=*/false, afrag, /*neg_b=*/false, bfrag,
        /*c_mod=*/(short)0, acc, /*reuse_a=*/false, /*reuse_b=*/false);
  }

  // --- (b) classifier sign, 8 (i,j) pairs per lane ------------------------
  const float b20 = b2[0], b21 = b2[1], b22 = b2[2];
#pragma unroll
  for (int r = 0; r < 8; ++r) {
    const int p  = lane + 32 * r;      // p = mi*16 + nj
    const int mi = p >> 4;
    const int nj = p & 15;
    const float4* __restrict__ ha = (const float4*)(sAi + mi * HID);
    const float4* __restrict__ hb = (const float4*)(sBj + nj * HID);
    const float4* __restrict__ hc = (const float4*)sB1;
    const float4* __restrict__ w0 = (const float4*)sW2;
    const float4* __restrict__ w1 = (const float4*)(sW2 + HID);
    const float4* __restrict__ w2 = (const float4*)(sW2 + 2 * HID);
    float l0 = b20, l1 = b21, l2 = b22;
#pragma unroll 4
    for (int k4 = 0; k4 < HID / 4; ++k4) {
      const float4 va = ha[k4], vb = hb[k4], vc = hc[k4];
      float4 h;
      h.x = fmaxf(va.x + vb.x + vc.x, 0.f);
      h.y = fmaxf(va.y + vb.y + vc.y, 0.f);
      h.z = fmaxf(va.z + vb.z + vc.z, 0.f);
      h.w = fmaxf(va.w + vb.w + vc.w, 0.f);
      const float4 u0 = w0[k4], u1 = w1[k4], u2 = w2[k4];
      l0 = fmaf(h.x, u0.x, fmaf(h.y, u0.y, fmaf(h.z, u0.z, fmaf(h.w, u0.w, l0))));
      l1 = fmaf(h.x, u1.x, fmaf(h.y, u1.y, fmaf(h.z, u1.z, fmaf(h.w, u1.w, l1))));
      l2 = fmaf(h.x, u2.x, fmaf(h.y, u2.y, fmaf(h.z, u2.z, fmaf(h.w, u2.w, l2))));
    }
    // torch argmax = first max: class0 wins ties vs {1,2}; class1 vs 2.
    const float s = (l0 >= l1 && l0 >= l2) ? 1.0f : ((l1 >= l2) ? 0.0f : -1.0f);
    sSign[p] = s;
  }
  __syncthreads();

  // --- (c) combine WMMA accumulator with signs, zero diagonal, store ------
  // C/D 16x16 f32 layout: VGPR v -> M = v + 8*(lane>=16), N = lane&15.
  const int n = lane & 15;
  const int mbase = (lane >> 4) << 3;
#pragma unroll
  for (int v = 0; v < 8; ++v) {
    const int m  = v + mbase;
    const int gi = i0 + m;
    const int gj = j0 + n;
    const float val = (gi == gj) ? 0.f : sSign[m * 16 + n] * acc[v];
    out[gi * N_GENES + gj] = val;
  }
}

// ---------------------------------------------------------------------------
extern "C" void kernel_launch(void* const* d_in, const int* in_sizes, int n_in,
                              void* d_out, int out_size, void* d_ws,
                              size_t ws_size, hipStream_t stream) {
  (void)in_sizes; (void)n_in; (void)out_size; (void)ws_size;
  const float* X  = (const float*)d_in[0];  // [768,128]
  const float* W1 = (const float*)d_in[1];  // [128,256]
  const float* b1 = (const float*)d_in[2];  // [128]
  const float* W2 = (const float*)d_in[3];  // [3,128]
  const float* b2 = (const float*)d_in[4];  // [3]
  const float* Wb = (const float*)d_in[5];  // [1,128,128]
  const float* bb = (const float*)d_in[6];  // [1]
  float* out = (float*)d_out;               // [768,768]

  float* ws = (float*)d_ws;                 // 3 * 768*128 floats = 1.18 MB
  float* Ai = ws;
  float* Bj = ws + N_GENES * HID;
  float* Y  = ws + 2 * N_GENES * HID;

  grn_proj_kernel<<<dim3(N_GENES), dim3(HID), 0, stream>>>(X, W1, Wb, Ai, Bj, Y);
  grn_tile_kernel<<<dim3(NTILE, NTILE), dim3(32), 0, stream>>>(
      X, Ai, Bj, Y, b1, W2, b2, bb, out);
}